// LSTM_ER_12695923327497
// MI455X (gfx1250) — compile-verified
//
#include <hip/hip_runtime.h>
#include <stdint.h>

#define B_ 32
#define T_ 2048
#define D_ 1024
#define U_ 32

typedef float v2f __attribute__((ext_vector_type(2)));
typedef float v4f __attribute__((ext_vector_type(4)));
typedef float v8f __attribute__((ext_vector_type(8)));

// ---------------------------------------------------------------------------
// Kernel 1: ragged span mean-pool via V_WMMA_F32_16X16X4_F32.
// One workgroup per (b,u). 16 waves; wave w owns columns [w*64, w*64+64).
// For each 4-timestep chunk: A = 16x4 indicator matrix (all rows equal,
// A[m,k] = 1 if t0+k < end), B = 4x16 slice of X rows t0..t0+3. C row 0
// accumulates sum over the span. Clamped row indices keep OOB reads safe;
// the indicator zeroes their contribution.
//
// A-layout (ISA 7.12.2, 32-bit A 16x4): lanes 0-15 -> K=0 (V0), K=1 (V1);
// lanes 16-31 -> K=2 (V0), K=3 (V1). B 4x16 mirrors the K split with
// N = lane&15. C: VGPR r, lanes 0-15 = row M=r -> row 0 is c[0], lanes 0-15.
// ---------------------------------------------------------------------------
__global__ __launch_bounds__(512) void span_mean_kernel(
    const float* __restrict__ X, const int* __restrict__ starts,
    const int* __restrict__ lens, float* __restrict__ mean)
{
  const int bu = blockIdx.x;              // b*U + u  (uniform)
  const int s  = starts[bu];
  const int L  = lens[bu] + 1;            // true span length (>=1)
  const int e  = s + L;                   // exclusive end
  const int b  = bu >> 5;

  const int lane = threadIdx.x & 31;
  const int wave = threadIdx.x >> 5;
  const int n    = lane & 15;
  const int koff = (lane >> 4) << 1;      // 0 for lanes 0-15, 2 for 16-31

  const float* Xb = X + (size_t)b * T_ * D_;
  const int col0 = wave * 64 + n;         // 4 tiles: col0, +16, +32, +48

  v8f c0 = {}; v8f c1 = {}; v8f c2 = {}; v8f c3 = {};

  for (int t0 = s; t0 < e; t0 += 4) {     // uniform trip count -> EXEC all 1s
    const int tx = t0 + koff;
    const int ty = tx + 1;
    v2f a;
    a.x = (tx < e) ? 1.0f : 0.0f;
    a.y = (ty < e) ? 1.0f : 0.0f;
    const int rx = (tx < T_ - 1) ? tx : (T_ - 1);   // clamp keeps loads legal
    const int ry = (ty < T_ - 1) ? ty : (T_ - 1);
    const float* px = Xb + (size_t)rx * D_;
    const float* py = Xb + (size_t)ry * D_;
    v2f b0 = { px[col0     ], py[col0     ] };
    v2f b1 = { px[col0 + 16], py[col0 + 16] };
    v2f b2 = { px[col0 + 32], py[col0 + 32] };
    v2f b3 = { px[col0 + 48], py[col0 + 48] };
    c0 = __builtin_amdgcn_wmma_f32_16x16x4_f32(false, a, false, b0, (short)0, c0, false, false);
    c1 = __builtin_amdgcn_wmma_f32_16x16x4_f32(false, a, false, b1, (short)0, c1, false, false);
    c2 = __builtin_amdgcn_wmma_f32_16x16x4_f32(false, a, false, b2, (short)0, c2, false, false);
    c3 = __builtin_amdgcn_wmma_f32_16x16x4_f32(false, a, false, b3, (short)0, c3, false, false);
  }

  const float rl = 1.0f / (float)L;
  if (lane < 16) {                        // row M=0 of C lives in lanes 0-15, VGPR 0
    float* m = mean + (size_t)bu * D_;
    m[col0     ] = c0[0] * rl;
    m[col0 + 16] = c1[0] * rl;
    m[col0 + 32] = c2[0] * rl;
    m[col0 + 48] = c3[0] * rl;
  }
}

// ---------------------------------------------------------------------------
// Kernel 2: pairwise expand. One workgroup per (b,i); stages the 32x1024
// mean block for batch b (128 KiB) into LDS with async B128 copies, then
// streams 32 output rows of 8 KiB with non-temporal B128 stores:
//   out[b, i*U+j, 0:D]   = mean[b,i]   (held in registers)
//   out[b, i*U+j, D:2D]  = mean[b,j]   (from LDS)
// ---------------------------------------------------------------------------
__global__ __launch_bounds__(256) void expand_kernel(
    const float* __restrict__ mean, float* __restrict__ out)
{
  const int bi  = blockIdx.x;             // b*U + i
  const int b   = bi >> 5;
  const int i   = bi & 31;
  const int tid = threadIdx.x;            // 256 threads; each owns one v4f/row

  extern __shared__ v4f sm[];             // 32 * 256 v4f = 128 KiB

  const v4f* mean4 = (const v4f*)mean;
  const v4f* mb    = mean4 + (size_t)b * U_ * 256;

  // Async-stage mean[b] into LDS (ASYNCcnt path).
  for (int k = 0; k < U_; ++k) {
    const int idx = k * 256 + tid;
    unsigned lds = (unsigned)(uintptr_t)(sm + idx);   // low 32 bits = LDS offset
    const v4f* g = mb + idx;
    asm volatile("global_load_async_to_lds_b128 %0, %1, off"
                 :: "v"(lds), "v"(g) : "memory");
  }
  asm volatile("s_wait_asynccnt 0" ::: "memory");
  __syncthreads();

  const v4f mi = sm[(i << 8) + tid];
  v4f* out4 = (v4f*)out + (size_t)bi * (U_ * 512);    // 512 v4f per out row

  for (int j = 0; j < U_; ++j) {
    v4f* row = out4 + (size_t)j * 512;
    __builtin_nontemporal_store(mi, row + tid);           // first half: mean[b,i]
    const v4f mj = sm[(j << 8) + tid];
    __builtin_nontemporal_store(mj, row + 256 + tid);     // second half: mean[b,j]
  }
}

// ---------------------------------------------------------------------------
extern "C" void kernel_launch(void* const* d_in, const int* in_sizes, int n_in,
                              void* d_out, int out_size, void* d_ws, size_t ws_size,
                              hipStream_t stream) {
  (void)in_sizes; (void)n_in; (void)out_size; (void)ws_size;
  const float* X      = (const float*)d_in[0];   // [B, T, D] f32
  const int*   starts = (const int*)d_in[1];     // [B, U] i32
  const int*   lens   = (const int*)d_in[2];     // [B, U] i32
  float*       mean   = (float*)d_ws;            // [B, U, D] f32 scratch (4 MiB)
  float*       out    = (float*)d_out;           // [B, U*U, 2D] f32

  span_mean_kernel<<<B_ * U_, 512, 0, stream>>>(X, starts, lens, mean);
  expand_kernel<<<B_ * U_, 256, (size_t)U_ * 256 * sizeof(v4f), stream>>>(mean, out);
}